// Hard_Extract_Weight_Sum_Cluster_64836826301212
// MI455X (gfx1250) — compile-verified
//
#include <hip/hip_runtime.h>
#include <hip/hip_bf16.h>
#include <float.h>
#include <limits.h>

// Problem constants (match reference)
#define B_DIM   32
#define S_DIM   512
#define D_DIM   768
#define HEADS   12
#define CLUSTER 5
#define INDEX_N 256
#define HEAD_N  (INDEX_N - CLUSTER)        // 251 rows from extract
#define HEAD_K  (INDEX_N - 1 - CLUSTER)    // 250 top indices
#define TAIL_N  (S_DIM - 1 - (INDEX_N - CLUSTER)) // 511 - 251 = 260 tail tokens
#define GROUP   53                          // ceil(265/5) tokens per cluster

typedef __attribute__((ext_vector_type(2))) float v2f;
typedef __attribute__((ext_vector_type(8))) float v8f;

// ---------------------------------------------------------------------------
// Stage 0: zero the column-sum accumulator
// ---------------------------------------------------------------------------
__global__ void zero_colsum_kernel(float* __restrict__ colsum) {
    int i = blockIdx.x * blockDim.x + threadIdx.x;
    if (i < B_DIM * S_DIM) colsum[i] = 0.0f;
}

// ---------------------------------------------------------------------------
// Stage 1: column sums of atten via V_WMMA_F32_16X16X4_F32 with A == ones.
// grid = (B, 4 col-blocks of 128, HEADS); block = 256 (8 waves).
// Each wave owns a 16-column strip of one head's 512x512 slice and streams
// 512 rows in 4-row tiles through the B operand; the f32 accumulator C
// gathers column sums (every row of D is identical). Memory-bound by design.
//
// Assumed f32 B 4x16 layout (mirror of the documented A 16x4 layout):
//   VGPR0: lanes 0-15 -> K=0,N=lane ; lanes 16-31 -> K=2,N=lane-16
//   VGPR1: lanes 0-15 -> K=1       ; lanes 16-31 -> K=3
// ---------------------------------------------------------------------------
__global__ void colsum_wmma_kernel(const float* __restrict__ atten,
                                   float* __restrict__ colsum) {
    const int b    = blockIdx.x;
    const int cb   = blockIdx.y;
    const int h    = blockIdx.z;
    const int wave = threadIdx.x >> 5;
    const int lane = threadIdx.x & 31;
    const int n     = lane & 15;
    const int khalf = lane >> 4;           // which K pair this lane-half holds

    const int j0 = cb * 128 + wave * 16;   // base column of this wave's strip
    const float* base = atten + (size_t)(b * HEADS + h) * S_DIM * S_DIM;

    v2f a; a.x = 1.0f; a.y = 1.0f;         // A = ones(16x4)
    v8f c = {};                             // f32 accumulator

    for (int i = 0; i < S_DIM; i += 4) {
        const float* rp = base + (size_t)(i + 2 * khalf) * S_DIM + j0 + n;
        v2f bb;
        bb.x = rp[0];        // K = 2*khalf,     N = n
        bb.y = rp[S_DIM];    // K = 2*khalf + 1, N = n
        c = __builtin_amdgcn_wmma_f32_16x16x4_f32(
                false, a, false, bb, (short)0, c, false, false);
    }
    // D row M=0 lives in c[0], lanes 0-15 (N = lane)
    if (lane < 16) {
        atomicAdd(&colsum[b * S_DIM + j0 + lane], c[0]);
    }
}

// ---------------------------------------------------------------------------
// Stage 2: attended[b][j] = (colsum - sum_h diag) / HEADS
// grid = (B); block = 512
// ---------------------------------------------------------------------------
__global__ void finalize_attended_kernel(const float* __restrict__ atten,
                                         const float* __restrict__ colsum,
                                         float* __restrict__ attended) {
    const int b = blockIdx.x;
    const int j = threadIdx.x;
    float diag = 0.0f;
    for (int h = 0; h < HEADS; ++h)
        diag += atten[((size_t)(b * HEADS + h) * S_DIM + j) * S_DIM + j];
    attended[b * S_DIM + j] = (colsum[b * S_DIM + j] - diag) * (1.0f / HEADS);
}

// ---------------------------------------------------------------------------
// Stage 3: per-batch rank/select/softmax. One block of 512 threads per batch.
//   * bitonic sort 512 (value desc) over tokens 1..511 (+ sentinel)
//   * ranks 0..249   -> head indices (+1), sorted ascending, CLS prepended
//   * ranks 251..510 -> tail (index+1, softmax weight), sorted by index asc
// ---------------------------------------------------------------------------
__global__ void select_sort_kernel(const float* __restrict__ attended,
                                   int* __restrict__ head_idx,
                                   int* __restrict__ tail_idx,
                                   float* __restrict__ tail_w) {
    __shared__ float sval[512];
    __shared__ int   sidx[512];
    __shared__ float sw[512];
    __shared__ float red[512];
    __shared__ int   hkey[512];
    __shared__ int   tkey[512];
    __shared__ float tval[512];

    const int b   = blockIdx.x;
    const int tid = threadIdx.x;

    // Load scores for tokens 1..511 (entry e corresponds to token e+1)
    if (tid < S_DIM - 1) {
        sval[tid] = attended[b * S_DIM + tid + 1];
        sidx[tid] = tid;
    } else {
        sval[tid] = -FLT_MAX;   // sentinel sorts last under descending order
        sidx[tid] = S_DIM - 1;
    }
    __syncthreads();

    // Bitonic sort: sval descending, carry sidx
    for (int k = 2; k <= 512; k <<= 1) {
        for (int j = k >> 1; j > 0; j >>= 1) {
            int p = tid ^ j;
            if (p > tid) {
                bool up = ((tid & k) == 0);
                float va = sval[tid], vb = sval[p];
                // descending overall: in "up" blocks bigger value goes first
                bool doswap = up ? (va < vb) : (va > vb);
                if (doswap) {
                    sval[tid] = vb; sval[p] = va;
                    int ia = sidx[tid]; sidx[tid] = sidx[p]; sidx[p] = ia;
                }
            }
            __syncthreads();
        }
    }

    // Softmax over tail ranks [HEAD_K+1, 511): max is the first tail value
    const float maxv = sval[HEAD_K + 1];
    float e = 0.0f;
    if (tid >= HEAD_K + 1 && tid < S_DIM - 1) e = __expf(sval[tid] - maxv);
    sw[tid]  = e;
    red[tid] = e;
    __syncthreads();
    for (int off = 256; off > 0; off >>= 1) {
        if (tid < off) red[tid] += red[tid + off];
        __syncthreads();
    }
    const float inv_sum = 1.0f / red[0];
    __syncthreads();

    // Head: indices(+1) of ranks 0..249, pad with INT_MAX, sort ascending
    hkey[tid] = (tid < HEAD_K) ? (sidx[tid] + 1) : INT_MAX;
    // Tail: pairs (index+1, weight) at ranks 251..510, pad with INT_MAX
    if (tid >= HEAD_K + 1 && tid < S_DIM - 1) {
        tkey[tid] = sidx[tid] + 1;
        tval[tid] = sw[tid] * inv_sum;
    } else {
        tkey[tid] = INT_MAX;
        tval[tid] = 0.0f;
    }
    __syncthreads();

    // Bitonic sort hkey ascending (no payload)
    for (int k = 2; k <= 512; k <<= 1) {
        for (int j = k >> 1; j > 0; j >>= 1) {
            int p = tid ^ j;
            if (p > tid) {
                bool up = ((tid & k) == 0);
                int ka = hkey[tid], kb = hkey[p];
                bool doswap = up ? (ka > kb) : (ka < kb);
                if (doswap) { hkey[tid] = kb; hkey[p] = ka; }
            }
            __syncthreads();
        }
    }

    // Bitonic sort tkey ascending, carry tval
    for (int k = 2; k <= 512; k <<= 1) {
        for (int j = k >> 1; j > 0; j >>= 1) {
            int p = tid ^ j;
            if (p > tid) {
                bool up = ((tid & k) == 0);
                int ka = tkey[tid], kb = tkey[p];
                bool doswap = up ? (ka > kb) : (ka < kb);
                if (doswap) {
                    tkey[tid] = kb; tkey[p] = ka;
                    float wa = tval[tid]; tval[tid] = tval[p]; tval[p] = wa;
                }
            }
            __syncthreads();
        }
    }

    // Emit
    if (tid == 0) head_idx[b * HEAD_N] = 0;                 // CLS token
    if (tid < HEAD_K) head_idx[b * HEAD_N + 1 + tid] = hkey[tid];
    if (tid < TAIL_N) {
        tail_idx[b * TAIL_N + tid] = tkey[tid];
        tail_w[b * TAIL_N + tid]   = tval[tid];
    }
}

// ---------------------------------------------------------------------------
// Stage 4a: gather head rows.  grid = (B, HEAD_N); block = 256
// ---------------------------------------------------------------------------
__global__ void gather_head_kernel(const float* __restrict__ x,
                                   const int* __restrict__ head_idx,
                                   float* __restrict__ out) {
    const int b = blockIdx.x, m = blockIdx.y;
    const int tok = head_idx[b * HEAD_N + m];
    const float* src = x + ((size_t)b * S_DIM + tok) * D_DIM;
    float* dst = out + ((size_t)b * INDEX_N + m) * D_DIM;
    for (int d = threadIdx.x; d < D_DIM; d += 256) dst[d] = src[d];
}

// ---------------------------------------------------------------------------
// Stage 4b: weighted cluster means.  grid = (B, CLUSTER); block = 256
// cluster c = mean over GROUP=53 consecutive index-sorted tail tokens
// (zero padding counted in the divisor, as in the reference)
// ---------------------------------------------------------------------------
__global__ void cluster_kernel(const float* __restrict__ x,
                               const int* __restrict__ tail_idx,
                               const float* __restrict__ tail_w,
                               float* __restrict__ out) {
    const int b = blockIdx.x, c = blockIdx.y;
    const int t0 = c * GROUP;
    const int t1 = (t0 + GROUP < TAIL_N) ? (t0 + GROUP) : TAIL_N;
    float acc0 = 0.0f, acc1 = 0.0f, acc2 = 0.0f;
    for (int t = t0; t < t1; ++t) {
        const int tok = tail_idx[b * TAIL_N + t];
        const float w = tail_w[b * TAIL_N + t];
        const float* src = x + ((size_t)b * S_DIM + tok) * D_DIM;
        acc0 += w * src[threadIdx.x];
        acc1 += w * src[threadIdx.x + 256];
        acc2 += w * src[threadIdx.x + 512];
    }
    float* dst = out + ((size_t)b * INDEX_N + HEAD_N + c) * D_DIM;
    const float inv = 1.0f / (float)GROUP;
    dst[threadIdx.x]       = acc0 * inv;
    dst[threadIdx.x + 256] = acc1 * inv;
    dst[threadIdx.x + 512] = acc2 * inv;
}

// ---------------------------------------------------------------------------
extern "C" void kernel_launch(void* const* d_in, const int* in_sizes, int n_in,
                              void* d_out, int out_size, void* d_ws, size_t ws_size,
                              hipStream_t stream) {
    const float* x     = (const float*)d_in[0];   // (32, 512, 768)
    const float* atten = (const float*)d_in[1];   // (384, 512, 512)
    float* out = (float*)d_out;                   // (32, 256, 768)

    // Workspace carve-up (~230 KB)
    float* colsum   = (float*)d_ws;                         // 32*512
    float* attended = colsum + B_DIM * S_DIM;               // 32*512
    int*   head_idx = (int*)(attended + B_DIM * S_DIM);     // 32*251
    int*   tail_idx = head_idx + B_DIM * HEAD_N;            // 32*260
    float* tail_w   = (float*)(tail_idx + B_DIM * TAIL_N);  // 32*260

    // Stage 0: zero accumulators
    zero_colsum_kernel<<<(B_DIM * S_DIM + 255) / 256, 256, 0, stream>>>(colsum);

    // Stage 1: WMMA column-sum reduction of atten (bandwidth-bound streaming)
    dim3 g1(B_DIM, S_DIM / 128, HEADS);
    colsum_wmma_kernel<<<g1, 256, 0, stream>>>(atten, colsum);

    // Stage 2: subtract diagonal, scale 1/HEADS
    finalize_attended_kernel<<<B_DIM, S_DIM, 0, stream>>>(atten, colsum, attended);

    // Stage 3: per-batch sort / select / softmax
    select_sort_kernel<<<B_DIM, 512, 0, stream>>>(attended, head_idx, tail_idx, tail_w);

    // Stage 4: gather + clusters
    dim3 g4a(B_DIM, HEAD_N);
    gather_head_kernel<<<g4a, 256, 0, stream>>>(x, head_idx, out);
    dim3 g4b(B_DIM, CLUSTER);
    cluster_kernel<<<g4b, 256, 0, stream>>>(x, tail_idx, tail_w, out);
}